// RoIAlign_56513179681073
// MI455X (gfx1250) — compile-verified
//
#include <hip/hip_runtime.h>
#include <hip/hip_bf16.h>

// RoIAlign as two dense f32 WMMA matmuls per (roi, channel):
//   out(7x7) = Wy(7x64) @ I_c(64x64) @ Wx^T(64x7)
// with the 2x2 sampling average folded into Wx / Wy (exact refactoring).

typedef __attribute__((ext_vector_type(2))) float v2f;
typedef __attribute__((ext_vector_type(8))) float v8f;

#define ROI_C   256
#define ROI_HW  64
#define ROI_PH  7
#define ROI_PW  7

__global__ __launch_bounds__(256) void roialign_wmma_kernel(
    const float* __restrict__ img,    // [B, 256, 64, 64]
    const float* __restrict__ rois,   // [K, 5]
    float* __restrict__ out,          // [K, 256, 7, 7]
    int K)
{
    __shared__ __align__(16) float sWx[16 * 64];     // padded 16x64 (rows 7..15 zero)
    __shared__ __align__(16) float sWy[16 * 64];
    __shared__ __align__(16) float sTmp[8][64 * 16]; // per-wave staging for tmp = I @ Wx^T
    __shared__ int sBatch;

    const int k    = blockIdx.x >> 5;   // roi index
    const int cg   = blockIdx.x & 31;   // channel octet
    const int tid  = threadIdx.x;
    const int wave = tid >> 5;
    const int lane = tid & 31;

    // ---- zero padded weight matrices ----
    #pragma unroll
    for (int i = tid; i < 16 * 64; i += 256) { sWx[i] = 0.0f; sWy[i] = 0.0f; }
    if (tid == 0) sBatch = (int)rois[(size_t)k * 5 + 0];
    __syncthreads();

    // ---- build axis weight matrices (aligned=true, sr=2, 1/4 folded as 1/2 * 1/2) ----
    if (tid < 14) {
        const int p   = tid % 7;          // pooled bin index along this axis
        const int isY = tid / 7;
        const float x1 = rois[(size_t)k * 5 + 1];
        const float y1 = rois[(size_t)k * 5 + 2];
        const float x2 = rois[(size_t)k * 5 + 3];
        const float y2 = rois[(size_t)k * 5 + 4];
        const float start = (isY ? y1 : x1) - 0.5f;
        const float len   = (isY ? y2 : x2) - 0.5f - start;
        const float bin   = len * (1.0f / 7.0f);
        float* Wrow = (isY ? sWy : sWx) + p * 64;
        const int lim = ROI_HW - 1;
        #pragma unroll
        for (int s = 0; s < 2; ++s) {
            float coord = start + (float)p * bin + ((float)s + 0.5f) * (bin * 0.5f);
            coord = fmaxf(coord, 0.0f);
            int lo = (int)coord;                       // trunc == floor (coord >= 0)
            int hi = (lo >= lim) ? lim : lo + 1;
            lo = min(lo, lim);
            const float lw = coord - (float)lo;
            const float hw = 1.0f - lw;
            Wrow[lo] += 0.5f * hw;                     // row-exclusive: no races
            Wrow[hi] += 0.5f * lw;
        }
    }
    __syncthreads();

    const int lrow    = lane & 15;
    const int rowHalf = lane >> 4;        // 0: K/M 0..1 half, 1: K/M 2..3 half
    const int off2    = rowHalf * 2;

    // ---- preload B operand (Wx) and A operand (Wy) into registers ----
    // B chunk kk: lane n holds Wx[n, 4kk+off2], Wx[n, 4kk+off2+1]
    // A chunk kk: lane m holds Wy[m, 4kk+off2], Wy[m, 4kk+off2+1]
    v2f bwx[16], awy[16];
    #pragma unroll
    for (int kk = 0; kk < 16; ++kk) {
        bwx[kk] = *reinterpret_cast<const v2f*>(&sWx[lrow * 64 + kk * 4 + off2]);
        awy[kk] = *reinterpret_cast<const v2f*>(&sWy[lrow * 64 + kk * 4 + off2]);
    }

    // ---- matmul 1: tmp(64x16) = I_c(64x64) @ Wx^T(64x16) ----
    const int c = cg * 8 + wave;
    const float* Ic = img + ((size_t)(sBatch * ROI_C + c)) * (ROI_HW * ROI_HW);

    v8f acc[4] = {v8f{}, v8f{}, v8f{}, v8f{}};
    #pragma unroll
    for (int mt = 0; mt < 4; ++mt) {
        const float* rowp = Ic + (size_t)(mt * 16 + lrow) * 64 + off2;
        #pragma unroll
        for (int kk = 0; kk < 16; ++kk) {
            const v2f a = *reinterpret_cast<const v2f*>(rowp + kk * 4); // global_load_b64
            acc[mt] = __builtin_amdgcn_wmma_f32_16x16x4_f32(
                false, a, false, bwx[kk], (short)0, acc[mt], false, false);
        }
    }

    // ---- D-layout -> B-layout transpose through per-wave LDS ----
    float* tp = sTmp[wave];
    #pragma unroll
    for (int mt = 0; mt < 4; ++mt) {
        #pragma unroll
        for (int v = 0; v < 8; ++v)
            tp[(mt * 16 + v + 8 * rowHalf) * 16 + lrow] = acc[mt][v];
    }

    // ---- matmul 2: out(16x16) = Wy(16x64) @ tmp(64x16) ----
    v8f acc2 = v8f{};
    #pragma unroll
    for (int kk = 0; kk < 16; ++kk) {
        v2f b;
        b.x = tp[(kk * 4 + off2    ) * 16 + lrow];
        b.y = tp[(kk * 4 + off2 + 1) * 16 + lrow];
        acc2 = __builtin_amdgcn_wmma_f32_16x16x4_f32(
            false, awy[kk], false, b, (short)0, acc2, false, false);
    }

    // ---- store 7x7: lane = pw (0..6), VGPR v = ph (0..6); rowHalf==0 only ----
    if (lane < ROI_PW) {
        float* op = out + (((size_t)k * ROI_C + c) * (ROI_PH * ROI_PW)) + lane;
        #pragma unroll
        for (int v = 0; v < ROI_PH; ++v)
            op[v * ROI_PW] = acc2[v];
    }
}

extern "C" void kernel_launch(void* const* d_in, const int* in_sizes, int n_in,
                              void* d_out, int out_size, void* d_ws, size_t ws_size,
                              hipStream_t stream) {
    const float* images = (const float*)d_in[0];
    const float* rois   = (const float*)d_in[1];
    float* out          = (float*)d_out;
    const int K = in_sizes[1] / 5;   // 512

    dim3 grid((unsigned)(K * 32));   // 32 channel-octets per roi
    dim3 block(256);                 // 8 wave32 waves = 8 channels
    roialign_wmma_kernel<<<grid, block, 0, stream>>>(images, rois, out, K);
}